// ProductQuantizer_62294205662006
// MI455X (gfx1250) — compile-verified
//
#include <hip/hip_runtime.h>
#include <hip/hip_bf16.h>

// ---------------------------------------------------------------------------
// Product quantizer for MI455X (gfx1250), wave32.
//   - distances via v_wmma_f32_16x16x4_f32 (exact f32, e_dim=64 -> 16 K-steps)
//   - codebook tiles staged into LDS by the Tensor Data Mover
//     (tensor_load_to_lds, double buffered, s_wait_tensorcnt), with TDM
//     padding (row stride 68 floats) for bank-conflict-free ds_load_b64.
//
// Inputs : d_in[0] = z         [16,512,32,32] f32
//          d_in[1] = codebooks [8,1024,64]    f32
// Output : d_out (f32, flat concat):
//          [0 .. 8388607]            z_vq  [16,512,32,32]
//          [8388608]                 loss  (scalar)
//          [8388609 .. 8519680]      min_encoding_indices [8*16384] (as f32)
//          [8519681 .. 8520704]      bin_count [1024] (as f32)
// Scratch: d_ws : int idx[8*16384] (512 KB)
// ---------------------------------------------------------------------------

typedef __attribute__((ext_vector_type(2))) float         v2f;
typedef __attribute__((ext_vector_type(8))) float         v8f;
typedef __attribute__((ext_vector_type(4))) unsigned int  u32x4;
typedef __attribute__((ext_vector_type(8))) int           i32x8;
typedef __attribute__((ext_vector_type(4))) int           i32x4;

#define PQ_M       8
#define PQ_NE      1024
#define PQ_EDIM    64
#define PQ_NPTS    16384          // 16*32*32 points
#define PQ_SPATIAL 1024           // 32*32 per batch image
#define PQ_CSTRIDE 1024           // channel stride in z (h*w)
#define PQ_BSTRIDE 524288         // 512*32*32 per batch image
#define PQ_MSTRIDE 65536          // 64*1024: sub-codebook channel-block stride in z
#define PQ_CBSTR   65536          // 1024*64: one sub-codebook

#define Z_ELEMS    8388608
#define LOSS_OFF   8388608
#define IDX_OFF    8388609
#define BIN_OFF    8519681        // 8388609 + 131072

#define LOSS_SCALE (1.25f / (16384.0f * 64.0f))   // (1+beta)/(N*e_dim)

#define LDS_ROW    68             // 64 data floats + 4 pad floats (TDM padding)
#define LDS_TILE   (16 * LDS_ROW) // 1088 floats per staged tile

// ---------------------------------------------------------------------------
// Kernel 0: zero loss + bin_count accumulators (d_out is poisoned, not zeroed)
// ---------------------------------------------------------------------------
__global__ __launch_bounds__(256)
void pq_init_kernel(float* __restrict__ out) {
    int t = blockIdx.x * blockDim.x + threadIdx.x;
    if (t == 0) out[LOSS_OFF] = 0.0f;
    if (t < PQ_NE) out[BIN_OFF + t] = 0.0f;
}

// ---------------------------------------------------------------------------
// TDM: stage one 16x64 f32 codebook tile (rows ct*16..ct*16+15) into LDS.
// Tensor = [1024 rows x 64 cols] f32, dim0_stride = 64.
// LDS padding: +4 DWORDs after every 64 DWORDs -> row stride 68 floats, so
// lane l16 reading row l16 at dword (l16*68 + off) hits bank (l16*4+off)%64:
// all 32 lanes conflict-free.
// ---------------------------------------------------------------------------
__device__ inline void tdm_load_tile(const float* gsrc, unsigned lds_off) {
    unsigned long long ga = (unsigned long long)(uintptr_t)gsrc;
    u32x4 g0 = { 0x1u,                                  // count=1 (valid user D#)
                 lds_off,                               // lds_addr (bytes)
                 (unsigned)ga,                          // global_addr[31:0]
                 (unsigned)(ga >> 32) | 0x80000000u };  // global_addr[56:32] | type=2
    i32x8 g1 = { 0x07520000,       // data_size=4B | pad_enable | interval=64dw | amount=4dw
                 64 << 16,         // tensor_dim0 = 64
                 1024 << 16,       // tensor_dim1 = 1024
                 64 << 16,         // tile_dim0 = 64
                 16,               // tile_dim1 = 16 (tile_dim2 = 0)
                 64,               // tensor_dim0_stride = 64
                 0, 0 };
    i32x4 gz4 = { 0, 0, 0, 0 };
    i32x8 gz8 = { 0, 0, 0, 0, 0, 0, 0, 0 };
    // 6-arg toolchain form: (g0, g1, g2, g3, extra, cpol)
    __builtin_amdgcn_tensor_load_to_lds(g0, g1, gz4, gz4, gz8, 0);
}

// ---------------------------------------------------------------------------
// Kernel 1: WMMA distance argmin.
// One wave per (sub-codebook m, 16-point tile); 8 waves/block share m and the
// B-tile stream, which the TDM double-buffers through LDS.
// score = ||e||^2 - 2*(z.e), argmin over 1024 codes.
// ---------------------------------------------------------------------------
__global__ __launch_bounds__(256)
void pq_argmin_kernel(const float* __restrict__ z,
                      const float* __restrict__ cb,
                      int* __restrict__ idx_out) {
    __shared__ float bstage[2][LDS_TILE];

    const int warp = threadIdx.x >> 5;
    const int wave = blockIdx.x * 8 + warp;             // 0..8191
    const int lane = threadIdx.x & 31;
    const int half = lane >> 4;       // 0: K pair {0,1}; 1: K pair {2,3}
    const int l16  = lane & 15;

    const int m    = wave >> 10;      // 0..7  (8 waves/block -> same m per block)
    const int tile = wave & 1023;     // 0..1023 (16-point tiles)
    const int b    = tile >> 6;       // 64 tiles per batch image
    const int sp0  = (tile & 63) << 4;

    const float* zb  = z  + b * PQ_BSTRIDE + m * PQ_MSTRIDE + sp0 + l16;
    const float* cbm = cb + m * PQ_CBSTR;

    const unsigned lds_base = (unsigned)(uintptr_t)&bstage[0][0];

    // --- A fragments: row M = l16, K(kk) = kk*4 + half*2 + {0,1} -------------
    float a[32];
#pragma unroll
    for (int kk = 0; kk < 16; ++kk) {
        const int k0 = kk * 4 + half * 2;
        a[2 * kk]     = zb[k0 * PQ_CSTRIDE];
        a[2 * kk + 1] = zb[(k0 + 1) * PQ_CSTRIDE];
    }

    float minv[8];
    int   mini[8];
#pragma unroll
    for (int j = 0; j < 8; ++j) { minv[j] = 3.4e38f; mini[j] = 0; }

    // Prime the pipeline: stage tile 0 into buffer 0 (wave 0 only).
    if (warp == 0) tdm_load_tile(cbm, lds_base);

    for (int ct = 0; ct < 64; ++ct) {
        const int buf = ct & 1;
        if (warp == 0) {
            if (ct + 1 < 64) {
                tdm_load_tile(cbm + (ct + 1) * 1024,
                              lds_base + ((ct + 1) & 1) * (LDS_TILE * 4));
                __builtin_amdgcn_s_wait_tensorcnt(1);   // tile ct resident
            } else {
                __builtin_amdgcn_s_wait_tensorcnt(0);
            }
        }
        __syncthreads();                                // tile ct visible to all

        const float* bs = &bstage[buf][0];
        const int code = ct * 16 + l16;                 // this lane's code column

        // --- B fragments from LDS + partial sum of squares ------------------
        float bb[32];
        float ss = 0.0f;
#pragma unroll
        for (int kk = 0; kk < 16; ++kk) {
            const float2 v =
                *(const float2*)(bs + l16 * LDS_ROW + kk * 4 + half * 2);
            bb[2 * kk]     = v.x;
            bb[2 * kk + 1] = v.y;
            ss = __builtin_fmaf(v.x, v.x, ss);
            ss = __builtin_fmaf(v.y, v.y, ss);
        }
        const float enorm = ss + __shfl_xor(ss, 16, 32);

        // --- z . e over K = 64 in 16 WMMA steps -----------------------------
        v8f acc = {};
#pragma unroll
        for (int kk = 0; kk < 16; ++kk) {
            v2f av; av.x = a[2 * kk];  av.y = a[2 * kk + 1];
            v2f bv; bv.x = bb[2 * kk]; bv.y = bb[2 * kk + 1];
            acc = __builtin_amdgcn_wmma_f32_16x16x4_f32(
                false, av, false, bv, (short)0, acc, false, false);
        }

        // --- running (min, argmin); rows M = j + 8*half per C layout --------
#pragma unroll
        for (int j = 0; j < 8; ++j) {
            const float s    = __builtin_fmaf(-2.0f, acc[j], enorm);
            const bool  take = s < minv[j];   // codes ascend -> first occurrence
            minv[j] = take ? s    : minv[j];
            mini[j] = take ? code : mini[j];
        }

        __syncthreads();            // done reading buf before it is re-staged
    }

    // --- reduce across the 16 code columns in each half-wave -----------------
#pragma unroll
    for (int off = 8; off >= 1; off >>= 1) {
#pragma unroll
        for (int j = 0; j < 8; ++j) {
            const float ov = __shfl_xor(minv[j], off, 32);
            const int   oi = __shfl_xor(mini[j], off, 32);
            const bool take = (ov < minv[j]) || (ov == minv[j] && oi < mini[j]);
            minv[j] = take ? ov : minv[j];
            mini[j] = take ? oi : mini[j];
        }
    }

    if (l16 == 0) {  // lane 0 -> rows 0..7, lane 16 -> rows 8..15
        int* o = idx_out + m * PQ_NPTS + tile * 16 + half * 8;
#pragma unroll
        for (int j = 0; j < 8; ++j) o[j] = mini[j];
    }
}

// ---------------------------------------------------------------------------
// Kernel 2: gather codes -> z_vq, loss partials, index tensor, bin counts.
// One thread per (m, n) pair (131072 threads), 64 dims each.
// ---------------------------------------------------------------------------
__global__ __launch_bounds__(256)
void pq_gather_kernel(const float* __restrict__ z,
                      const float* __restrict__ cb,
                      const int* __restrict__ idx_in,
                      float* __restrict__ out) {
    const int t = blockIdx.x * 256 + (int)threadIdx.x;   // 0..131071
    const int m = t >> 14;
    const int n = t & (PQ_NPTS - 1);
    const int b = n >> 10;
    const int sp = n & (PQ_SPATIAL - 1);

    const int code = idx_in[t];
    const float* crow = cb + m * PQ_CBSTR + code * PQ_EDIM;
    const float* zrow = z   + b * PQ_BSTRIDE + m * PQ_MSTRIDE + sp;
    float*       vrow = out + b * PQ_BSTRIDE + m * PQ_MSTRIDE + sp;

    float err = 0.0f;
#pragma unroll
    for (int d = 0; d < PQ_EDIM; ++d) {
        const float e  = crow[d];
        const float zf = zrow[d * PQ_CSTRIDE];
        vrow[d * PQ_CSTRIDE] = e;                 // z_vq forward value == zq
        const float df = e - zf;
        err = __builtin_fmaf(df, df, err);
    }

    out[IDX_OFF + t] = (float)code;
    atomicAdd(&out[BIN_OFF + code], 1.0f);

    // wave-level reduction of squared error, then one atomic per wave
#pragma unroll
    for (int off = 16; off >= 1; off >>= 1)
        err += __shfl_xor(err, off, 32);
    if ((threadIdx.x & 31) == 0)
        atomicAdd(&out[LOSS_OFF], err * LOSS_SCALE);
}

// ---------------------------------------------------------------------------
extern "C" void kernel_launch(void* const* d_in, const int* in_sizes, int n_in,
                              void* d_out, int out_size, void* d_ws, size_t ws_size,
                              hipStream_t stream) {
    const float* z  = (const float*)d_in[0];
    const float* cb = (const float*)d_in[1];
    float* out = (float*)d_out;
    int* idx_ws = (int*)d_ws;                       // 8*16384 ints = 512 KB

    pq_init_kernel<<<dim3((PQ_NE + 255) / 256), dim3(256), 0, stream>>>(out);
    pq_argmin_kernel<<<dim3(1024), dim3(256), 0, stream>>>(z, cb, idx_ws);
    pq_gather_kernel<<<dim3((PQ_M * PQ_NPTS) / 256), dim3(256), 0, stream>>>(
        z, cb, idx_ws, out);
}